// SPNNet_12068858101769
// MI455X (gfx1250) — compile-verified
//
#include <hip/hip_runtime.h>
#include <math.h>

typedef __attribute__((ext_vector_type(2))) float v2f;
typedef __attribute__((ext_vector_type(8))) float v8f;

#define BATCH 32768
#define DIN   512
#define HID   20
#define OUTN  5
#define ROWS_PER_BLOCK 128   // 8 waves * 16 rows
#define LOG_2PI 1.8378770664093453f

// ---------------------------------------------------------------------------
// Kernel 1: fused fc1(WMMA f32) + ReLU + SPN leaf/product/sum/root + log(-root)
// ---------------------------------------------------------------------------
__global__ __launch_bounds__(256) void spn_fused_kernel(
    const float* __restrict__ x,      // [B, 512]
    const float* __restrict__ W1,     // [20, 512]
    const float* __restrict__ b1,     // [20]
    const float* __restrict__ means0, // [5, 20]
    const float* __restrict__ stds0,  // [5, 20]
    const float* __restrict__ means1, // [5, 20]
    const float* __restrict__ stds1,  // [5, 20]
    const float* __restrict__ log_w,  // [5, 20, 20, 2]
    float* __restrict__ y_ws,         // [B, 5]
    float* __restrict__ blockmax)     // [gridDim.x]
{
    __shared__ float sh_h[ROWS_PER_BLOCK][HID];       // 10 KB
    __shared__ float sh_lw0[OUTN * HID * HID];        // 8 KB
    __shared__ float sh_lw1[OUTN * HID * HID];        // 8 KB
    __shared__ float sh_gm0[OUTN * HID], sh_gi0[OUTN * HID], sh_gc0[OUTN * HID];
    __shared__ float sh_gm1[OUTN * HID], sh_gi1[OUTN * HID], sh_gc1[OUTN * HID];
    __shared__ float sh_red[256];

    const int tid = threadIdx.x;

    // ---- Phase 0: stage SPN params in LDS -------------------------------
    if (tid < OUTN * HID) {
        float s0 = stds0[tid], s1 = stds1[tid];
        sh_gm0[tid] = means0[tid];
        sh_gm1[tid] = means1[tid];
        sh_gi0[tid] = 1.0f / s0;
        sh_gi1[tid] = 1.0f / s1;
        sh_gc0[tid] = -__logf(s0) - 0.5f * LOG_2PI;
        sh_gc1[tid] = -__logf(s1) - 0.5f * LOG_2PI;
    }
    for (int idx = tid; idx < OUTN * HID * HID; idx += 256) {
        sh_lw0[idx] = log_w[2 * idx + 0];
        sh_lw1[idx] = log_w[2 * idx + 1];
    }

    // ---- Phase 1: fc1 via V_WMMA_F32_16X16X4_F32 ------------------------
    // Each wave: 16 rows of x, 32 cols of h (cols 20..31 padded with zeros).
    const int wave = tid >> 5;
    const int lane = tid & 31;
    const int lh   = lane & 15;       // M (and N) index within half-wave
    const int kh   = (lane >> 4) * 2; // K sub-offset: lanes 0-15 -> K0,K1; 16-31 -> K2,K3
    const int m0   = blockIdx.x * ROWS_PER_BLOCK + wave * 16;

    const float* xrow  = x  + (size_t)(m0 + lh) * DIN + kh;
    const float* wrow0 = W1 + (size_t)lh * DIN + kh;             // N = lh   (< 20)
    const bool   n1ok  = (16 + lh) < HID;                        // lh < 4
    const float* wrow1 = W1 + (size_t)(n1ok ? (16 + lh) : 0) * DIN + kh;

    v8f c0 = {0.f,0.f,0.f,0.f,0.f,0.f,0.f,0.f};
    v8f c1 = c0;

    #pragma unroll 4
    for (int k = 0; k < DIN; k += 4) {
        v2f a  = *(const v2f*)(xrow  + k);
        v2f b0 = *(const v2f*)(wrow0 + k);
        v2f bb = *(const v2f*)(wrow1 + k);
        if (!n1ok) { bb.x = 0.f; bb.y = 0.f; }
        c0 = __builtin_amdgcn_wmma_f32_16x16x4_f32(false, a, false, b0,
                                                   (short)0, c0, false, false);
        c1 = __builtin_amdgcn_wmma_f32_16x16x4_f32(false, a, false, bb,
                                                   (short)0, c1, false, false);
    }

    // bias + ReLU + scatter h tile into LDS
    {
        const float bias0 = b1[lh];
        const float bias1 = n1ok ? b1[16 + lh] : 0.f;
        const int rowbase = wave * 16 + ((lane < 16) ? 0 : 8);
        #pragma unroll
        for (int r = 0; r < 8; ++r) {
            const int row = rowbase + r;           // M = r (or r+8)
            sh_h[row][lh] = fmaxf(c0[r] + bias0, 0.f);
            if (n1ok)
                sh_h[row][16 + lh] = fmaxf(c1[r] + bias1, 0.f);
        }
    }
    __syncthreads();

    // ---- Phase 2: SPN per (row, o) task ---------------------------------
    float localmax = -INFINITY;
    for (int task = tid; task < ROWS_PER_BLOCK * OUTN; task += 256) {
        const int lrow = task / OUTN;
        const int o    = task - lrow * OUTN;

        float L0[HID], L1[HID];
        #pragma unroll
        for (int i = 0; i < HID; ++i) {
            const float hv = sh_h[lrow][i];
            const float z0 = (hv - sh_gm0[o * HID + i]) * sh_gi0[o * HID + i];
            const float z1 = (hv - sh_gm1[o * HID + i]) * sh_gi1[o * HID + i];
            L0[i] = fmaf(-0.5f * z0, z0, sh_gc0[o * HID + i]);
            L1[i] = fmaf(-0.5f * z1, z1, sh_gc1[o * HID + i]);
        }

        const float* p0 = sh_lw0 + o * HID * HID;
        const float* p1 = sh_lw1 + o * HID * HID;
        float s = 0.f;
        for (int i = 0; i < HID; ++i) {
            const float a0 = L0[i], a1 = L1[i];
            #pragma unroll 5
            for (int j = 0; j < HID; ++j) {
                const float pa = p0[i * HID + j] + a0 + L0[j];
                const float pb = p1[i * HID + j] + a1 + L1[j];
                const float m  = fmaxf(pa, pb);
                const float d  = fminf(pa, pb) - m;      // -|pa - pb|
                s += m + __logf(1.0f + __expf(d));       // logaddexp
            }
        }

        const float yv = __logf(-s);                     // log(-root)
        y_ws[(size_t)(blockIdx.x * ROWS_PER_BLOCK + lrow) * OUTN + o] = yv;
        localmax = fmaxf(localmax, yv);
    }

    // block max reduction
    sh_red[tid] = localmax;
    __syncthreads();
    for (int sft = 128; sft > 0; sft >>= 1) {
        if (tid < sft) sh_red[tid] = fmaxf(sh_red[tid], sh_red[tid + sft]);
        __syncthreads();
    }
    if (tid == 0) blockmax[blockIdx.x] = sh_red[0];
}

// ---------------------------------------------------------------------------
// Kernel 2: reduce 256 block maxima -> global max
// ---------------------------------------------------------------------------
__global__ __launch_bounds__(256) void spn_max_kernel(const float* __restrict__ blockmax,
                                                      float* __restrict__ maxval)
{
    __shared__ float red[256];
    red[threadIdx.x] = blockmax[threadIdx.x];
    __syncthreads();
    for (int sft = 128; sft > 0; sft >>= 1) {
        if (threadIdx.x < sft)
            red[threadIdx.x] = fmaxf(red[threadIdx.x], red[threadIdx.x + sft]);
        __syncthreads();
    }
    if (threadIdx.x == 0) maxval[0] = red[0];
}

// ---------------------------------------------------------------------------
// Kernel 3: out[b] = sigmoid( sum_o (M - y[b,o]) * W2[o] + b2 )
// ---------------------------------------------------------------------------
__global__ __launch_bounds__(256) void spn_head_kernel(const float* __restrict__ y_ws,
                                                       const float* __restrict__ maxval,
                                                       const float* __restrict__ W2,
                                                       const float* __restrict__ b2,
                                                       float* __restrict__ out)
{
    const int b = blockIdx.x * blockDim.x + threadIdx.x;
    if (b >= BATCH) return;
    const float M = maxval[0];
    float acc = b2[0];
    #pragma unroll
    for (int o = 0; o < OUTN; ++o)
        acc = fmaf(M - y_ws[(size_t)b * OUTN + o], W2[o], acc);
    out[b] = 1.0f / (1.0f + __expf(-acc));
}

// ---------------------------------------------------------------------------
extern "C" void kernel_launch(void* const* d_in, const int* in_sizes, int n_in,
                              void* d_out, int out_size, void* d_ws, size_t ws_size,
                              hipStream_t stream)
{
    const float* x      = (const float*)d_in[0];
    const float* W1     = (const float*)d_in[1];
    const float* b1     = (const float*)d_in[2];
    const float* means0 = (const float*)d_in[3];
    const float* stds0  = (const float*)d_in[4];
    const float* means1 = (const float*)d_in[5];
    const float* stds1  = (const float*)d_in[6];
    const float* log_w  = (const float*)d_in[7];
    const float* W2     = (const float*)d_in[8];
    const float* b2     = (const float*)d_in[9];
    float* out = (float*)d_out;

    float* ws       = (float*)d_ws;
    float* y_ws     = ws;                      // BATCH*OUTN floats
    float* blockmax = ws + (size_t)BATCH * OUTN; // 256 floats
    float* maxval   = blockmax + 256;          // 1 float

    const int nblk = BATCH / ROWS_PER_BLOCK;   // 256
    spn_fused_kernel<<<nblk, 256, 0, stream>>>(x, W1, b1, means0, stds0,
                                               means1, stds1, log_w, y_ws, blockmax);
    spn_max_kernel<<<1, 256, 0, stream>>>(blockmax, maxval);
    spn_head_kernel<<<(BATCH + 255) / 256, 256, 0, stream>>>(y_ws, maxval, W2, b2, out);
}